// MPSWithDenseBond3_7224134992124
// MI455X (gfx1250) — compile-verified
//
#include <hip/hip_runtime.h>
#include <stdint.h>

// ---------------- problem constants ----------------
#define N_IN     128               // features per sample
#define THREADS  64                // threads per block (2 waves)
#define SPT      2                 // samples per thread (packed fp32)
#define ROWS     (THREADS * SPT)   // 128 sample rows staged per block
#define CHUNK    16                // features staged per chunk
#define ROW_DW   20                // 16 data floats + 4 pad dwords (80 B; 16B-aligned; <=2-way LDS conflicts)
#define BUF_DW   (ROWS * ROW_DW)   // one staging buffer: 2560 dwords = 10 KB (x2 double buffer)
#define NSITE    63                // interior sites per half-chain

typedef __attribute__((ext_vector_type(2))) float f32x2;

// Constant-AS views: force s_load through the scalar/constant cache even though
// the async-copy asm clobbers "memory" (ws is immutable during the main kernel).
typedef const __attribute__((address_space(4))) f32x2 cs_f32x2;
typedef const __attribute__((address_space(4))) float cs_float;

// ---- CDNA5 async global->LDS copy (ASYNCcnt path), 16 bytes per lane ----
__device__ __forceinline__ void async_ld16(unsigned lds_byte, const char* gaddr) {
  asm volatile("global_load_async_to_lds_b128 %0, %1, off"
               :: "v"(lds_byte), "v"(gaddr) : "memory");
}
__device__ __forceinline__ void wait_async_le8() {
  asm volatile("s_wait_asynccnt 8" ::: "memory");
}
__device__ __forceinline__ void wait_async_0() {
  asm volatile("s_wait_asynccnt 0" ::: "memory");
}

// ---- VOP3P packed fp32 (64-bit operands: OPSEL/OPSEL_HI encoded 0) ----
__device__ __forceinline__ f32x2 pk_mul_ws(f32x2 a, f32x2 w) {
  f32x2 d;
  asm("v_pk_mul_f32 %0, %1, %2 op_sel:[0,0] op_sel_hi:[0,0]"
      : "=v"(d) : "v"(a), "s"(w));
  return d;
}
__device__ __forceinline__ f32x2 pk_fma_ws(f32x2 a, f32x2 w, f32x2 c) {
  f32x2 d;
  asm("v_pk_fma_f32 %0, %1, %2, %3 op_sel:[0,0,0] op_sel_hi:[0,0,0]"
      : "=v"(d) : "v"(a), "s"(w), "v"(c));
  return d;
}
__device__ __forceinline__ f32x2 pk_fma_vv(f32x2 a, f32x2 b, f32x2 c) {
  f32x2 d;
  asm("v_pk_fma_f32 %0, %1, %2, %3 op_sel:[0,0,0] op_sel_hi:[0,0,0]"
      : "=v"(d) : "v"(a), "v"(b), "v"(c));
  return d;
}

// ---------------- setup kernel ----------------
// 1) Fold dense head (linear∘linear, no activation) + T_out into G[4][4], b_eff:
//      out = v^T G u + b_eff,  G[l][r] = sum_o T_out[l,o,r] * (W1@W2)[o]
// 2) Duplicate chain weights into {w,w} pairs so the main kernel feeds
//    v_pk_fma_f32 straight from aligned SGPR pairs (s_load of 256B/site).
// ws layout (floats): [0..16] G+b_eff, [64..] dupL (63*64), then dupR (63*64).
__global__ void setup_kernel(const float* __restrict__ T_out,  // [4][8][4]
                             const float* __restrict__ W1,     // [8][24]
                             const float* __restrict__ b1,     // [24]
                             const float* __restrict__ W2,     // [24][1]
                             const float* __restrict__ b2,     // [1]
                             const float* __restrict__ T_left, // [63][4][2][4]
                             const float* __restrict__ T_right,// [63][4][2][4]
                             float* __restrict__ ws) {
  const int t = threadIdx.x;
  if (t < 16) {
    float w[8];
#pragma unroll
    for (int o = 0; o < 8; ++o) {
      float s = 0.f;
#pragma unroll
      for (int k = 0; k < 24; ++k) s = fmaf(W1[o * 24 + k], W2[k], s);
      w[o] = s;
    }
    const int l = t >> 2, r = t & 3;
    float g = 0.f;
#pragma unroll
    for (int o = 0; o < 8; ++o) g = fmaf(T_out[l * 32 + o * 4 + r], w[o], g);
    ws[t] = g;
  } else if (t == 16) {
    float be = b2[0];
#pragma unroll
    for (int k = 0; k < 24; ++k) be = fmaf(b1[k], W2[k], be);
    ws[16] = be;
  }
  f32x2* dupL = (f32x2*)(ws + 64);
  f32x2* dupR = dupL + NSITE * 32;
  for (int i = t; i < NSITE * 32; i += blockDim.x) {
    f32x2 a; a.x = T_left[i];  a.y = T_left[i];  dupL[i] = a;
    f32x2 c; c.x = T_right[i]; c.y = T_right[i]; dupR[i] = c;
  }
}

// ---------------- main kernel: 2 samples/lane, packed fp32 chain ----------------
__global__ __launch_bounds__(THREADS) void mps_chain_kernel(
    const float* __restrict__ x,        // [BATCH][128]
    const float* __restrict__ T_first,  // [2][4]
    const float* __restrict__ T_last,   // [4][2]
    const float* __restrict__ ws,       // setup_kernel output
    float* __restrict__ out)            // [BATCH]
{
  __shared__ float smem[2 * BUF_DW];    // 20 KB

  const int t = threadIdx.x;
  cs_float* Gb   = (cs_float*)(uintptr_t)ws;
  cs_f32x2* dupL = (cs_f32x2*)(uintptr_t)(ws + 64);
  cs_f32x2* dupR = dupL + NSITE * 32;

  const char* xblk = (const char*)(x + (size_t)blockIdx.x * ROWS * N_IN);
  // low 32 bits of the generic LDS address == LDS byte offset (ISA 10.2 aperture rules)
  const unsigned lbase = (unsigned)(uintptr_t)(&smem[0]);

  // Stage chunk c (16 floats = 64 B of each of the 128 rows) into buffer `buf`.
  auto issue = [&](int c, int buf) {
    const char* g0 = xblk + c * (CHUNK * 4);
    const unsigned l0 = lbase + (unsigned)buf * (BUF_DW * 4);
#pragma unroll
    for (int j = 0; j < 8; ++j) {
      const int i = j * THREADS + t;
      const int r = i >> 2, s = i & 3;
      async_ld16(l0 + (unsigned)(r * (ROW_DW * 4) + s * 16),
                 g0 + (size_t)r * (N_IN * 4) + s * 16);
    }
  };

  f32x2 v0{}, v1{}, v2{}, v3{};   // left-chain state  (two samples packed)
  f32x2 u0{}, u1{}, u2{}, u3{};   // right-chain state

  // chunk order: 0,1,2,3 (v, forward) then 7,6,5,4 (u, reverse within chunk)
  issue(0, 0);
  for (int k = 0; k < 8; ++k) {
    const int c = (k < 4) ? k : (11 - k);
    const int buf = k & 1;
    if (k < 7) {
      const int kn = k + 1;
      issue((kn < 4) ? kn : (11 - kn), kn & 1);
      wait_async_le8();           // current chunk's 8 asyncs done (in-order completion)
    } else {
      wait_async_0();
    }
    __syncthreads();              // make all waves' staged data visible

    const float* rowA = &smem[buf * BUF_DW + (2 * t) * ROW_DW];
    const float* rowB = rowA + ROW_DW;

    if (k < 4) {
      // ----- left chain: v <- v@A_n + x .* (v@B_n) -----
      int kk0 = 0;
      if (k == 0) {
        const float xa = rowA[0], xb = rowB[0];   // feature 0 -> T_first init
        v0.x = fmaf(xa, T_first[4], T_first[0]);  v0.y = fmaf(xb, T_first[4], T_first[0]);
        v1.x = fmaf(xa, T_first[5], T_first[1]);  v1.y = fmaf(xb, T_first[5], T_first[1]);
        v2.x = fmaf(xa, T_first[6], T_first[2]);  v2.y = fmaf(xb, T_first[6], T_first[2]);
        v3.x = fmaf(xa, T_first[7], T_first[3]);  v3.y = fmaf(xb, T_first[7], T_first[3]);
        kk0 = 1;
      }
      for (int kk = kk0; kk < CHUNK; ++kk) {
        f32x2 xv; xv.x = rowA[kk]; xv.y = rowB[kk];
        cs_f32x2* W = &dupL[(size_t)(c * CHUNK + kk - 1) * 32];  // site f-1
        f32x2 a0 = pk_fma_ws(v3, W[24], pk_fma_ws(v2, W[16], pk_fma_ws(v1, W[ 8], pk_mul_ws(v0, W[ 0]))));
        f32x2 a1 = pk_fma_ws(v3, W[25], pk_fma_ws(v2, W[17], pk_fma_ws(v1, W[ 9], pk_mul_ws(v0, W[ 1]))));
        f32x2 a2 = pk_fma_ws(v3, W[26], pk_fma_ws(v2, W[18], pk_fma_ws(v1, W[10], pk_mul_ws(v0, W[ 2]))));
        f32x2 a3 = pk_fma_ws(v3, W[27], pk_fma_ws(v2, W[19], pk_fma_ws(v1, W[11], pk_mul_ws(v0, W[ 3]))));
        f32x2 e0 = pk_fma_ws(v3, W[28], pk_fma_ws(v2, W[20], pk_fma_ws(v1, W[12], pk_mul_ws(v0, W[ 4]))));
        f32x2 e1 = pk_fma_ws(v3, W[29], pk_fma_ws(v2, W[21], pk_fma_ws(v1, W[13], pk_mul_ws(v0, W[ 5]))));
        f32x2 e2 = pk_fma_ws(v3, W[30], pk_fma_ws(v2, W[22], pk_fma_ws(v1, W[14], pk_mul_ws(v0, W[ 6]))));
        f32x2 e3 = pk_fma_ws(v3, W[31], pk_fma_ws(v2, W[23], pk_fma_ws(v1, W[15], pk_mul_ws(v0, W[ 7]))));
        v0 = pk_fma_vv(xv, e0, a0);
        v1 = pk_fma_vv(xv, e1, a1);
        v2 = pk_fma_vv(xv, e2, a2);
        v3 = pk_fma_vv(xv, e3, a3);
      }
    } else {
      // ----- right chain (reverse): u <- A_n@u + x .* (B_n@u) -----
      int kk0 = CHUNK - 1;
      if (k == 4) {
        const float xa = rowA[CHUNK - 1], xb = rowB[CHUNK - 1];  // feature 127 -> T_last init
        u0.x = fmaf(xa, T_last[1], T_last[0]);  u0.y = fmaf(xb, T_last[1], T_last[0]);
        u1.x = fmaf(xa, T_last[3], T_last[2]);  u1.y = fmaf(xb, T_last[3], T_last[2]);
        u2.x = fmaf(xa, T_last[5], T_last[4]);  u2.y = fmaf(xb, T_last[5], T_last[4]);
        u3.x = fmaf(xa, T_last[7], T_last[6]);  u3.y = fmaf(xb, T_last[7], T_last[6]);
        kk0 = CHUNK - 2;
      }
      for (int kk = kk0; kk >= 0; --kk) {
        f32x2 xv; xv.x = rowA[kk]; xv.y = rowB[kk];
        cs_f32x2* W = &dupR[(size_t)(c * CHUNK + kk - 64) * 32];  // site f-64
        f32x2 a0 = pk_fma_ws(u3, W[ 3], pk_fma_ws(u2, W[ 2], pk_fma_ws(u1, W[ 1], pk_mul_ws(u0, W[ 0]))));
        f32x2 e0 = pk_fma_ws(u3, W[ 7], pk_fma_ws(u2, W[ 6], pk_fma_ws(u1, W[ 5], pk_mul_ws(u0, W[ 4]))));
        f32x2 a1 = pk_fma_ws(u3, W[11], pk_fma_ws(u2, W[10], pk_fma_ws(u1, W[ 9], pk_mul_ws(u0, W[ 8]))));
        f32x2 e1 = pk_fma_ws(u3, W[15], pk_fma_ws(u2, W[14], pk_fma_ws(u1, W[13], pk_mul_ws(u0, W[12]))));
        f32x2 a2 = pk_fma_ws(u3, W[19], pk_fma_ws(u2, W[18], pk_fma_ws(u1, W[17], pk_mul_ws(u0, W[16]))));
        f32x2 e2 = pk_fma_ws(u3, W[23], pk_fma_ws(u2, W[22], pk_fma_ws(u1, W[21], pk_mul_ws(u0, W[20]))));
        f32x2 a3 = pk_fma_ws(u3, W[27], pk_fma_ws(u2, W[26], pk_fma_ws(u1, W[25], pk_mul_ws(u0, W[24]))));
        f32x2 e3 = pk_fma_ws(u3, W[31], pk_fma_ws(u2, W[30], pk_fma_ws(u1, W[29], pk_mul_ws(u0, W[28]))));
        u0 = pk_fma_vv(xv, e0, a0);
        u1 = pk_fma_vv(xv, e1, a1);
        u2 = pk_fma_vv(xv, e2, a2);
        u3 = pk_fma_vv(xv, e3, a3);
      }
    }
    __syncthreads();              // all waves done reading buf before it is re-staged
  }

  // ----- out = v^T G u + b_eff (unpacked, once per sample) -----
  const float t0a = fmaf(v3.x, Gb[12], fmaf(v2.x, Gb[ 8], fmaf(v1.x, Gb[ 4], v0.x * Gb[ 0])));
  const float t1a = fmaf(v3.x, Gb[13], fmaf(v2.x, Gb[ 9], fmaf(v1.x, Gb[ 5], v0.x * Gb[ 1])));
  const float t2a = fmaf(v3.x, Gb[14], fmaf(v2.x, Gb[10], fmaf(v1.x, Gb[ 6], v0.x * Gb[ 2])));
  const float t3a = fmaf(v3.x, Gb[15], fmaf(v2.x, Gb[11], fmaf(v1.x, Gb[ 7], v0.x * Gb[ 3])));
  const float rA  = fmaf(t3a, u3.x, fmaf(t2a, u2.x, fmaf(t1a, u1.x, fmaf(t0a, u0.x, Gb[16]))));

  const float t0b = fmaf(v3.y, Gb[12], fmaf(v2.y, Gb[ 8], fmaf(v1.y, Gb[ 4], v0.y * Gb[ 0])));
  const float t1b = fmaf(v3.y, Gb[13], fmaf(v2.y, Gb[ 9], fmaf(v1.y, Gb[ 5], v0.y * Gb[ 1])));
  const float t2b = fmaf(v3.y, Gb[14], fmaf(v2.y, Gb[10], fmaf(v1.y, Gb[ 6], v0.y * Gb[ 2])));
  const float t3b = fmaf(v3.y, Gb[15], fmaf(v2.y, Gb[11], fmaf(v1.y, Gb[ 7], v0.y * Gb[ 3])));
  const float rB  = fmaf(t3b, u3.y, fmaf(t2b, u2.y, fmaf(t1b, u1.y, fmaf(t0b, u0.y, Gb[16]))));

  f32x2 r; r.x = rA; r.y = rB;
  *(f32x2*)(out + (size_t)blockIdx.x * ROWS + 2 * t) = r;   // contiguous pair, one b64 store
}

// ---------------- host-side launch ----------------
extern "C" void kernel_launch(void* const* d_in, const int* in_sizes, int n_in,
                              void* d_out, int out_size, void* d_ws, size_t ws_size,
                              hipStream_t stream) {
  (void)n_in; (void)out_size; (void)ws_size;
  const float* x  = (const float*)d_in[0];  // [BATCH,128]
  const float* Tf = (const float*)d_in[1];  // T_first [2,4]
  const float* TL = (const float*)d_in[2];  // T_left  [63,4,2,4]
  const float* To = (const float*)d_in[3];  // T_out   [4,8,4]
  const float* TR = (const float*)d_in[4];  // T_right [63,4,2,4]
  const float* Tl = (const float*)d_in[5];  // T_last  [4,2]
  const float* W1 = (const float*)d_in[6];  // [8,24]
  const float* b1 = (const float*)d_in[7];  // [24]
  const float* W2 = (const float*)d_in[8];  // [24,1]
  const float* b2 = (const float*)d_in[9];  // [1]
  float* ws  = (float*)d_ws;                // ~32 KB scratch (G/b_eff + duplicated weights)
  float* out = (float*)d_out;

  const int batch = in_sizes[0] / N_IN;
  setup_kernel<<<1, 256, 0, stream>>>(To, W1, b1, W2, b2, TL, TR, ws);
  mps_chain_kernel<<<batch / ROWS, THREADS, 0, stream>>>(x, Tf, Tl, ws, out);
}